// GraphSAGE_34419867910942
// MI455X (gfx1250) — compile-verified
//
#include <hip/hip_runtime.h>

// ---------------------------------------------------------------------------
// GraphSAGE (3x SAGEConv + BN + ReLU) for MI455X / gfx1250, wave32.
//  Phase A (bandwidth-bound): degree + scatter-mean via global FP32 atomics.
//  Phase B (compute, fused):  D = (agg/deg) @ Wl^T + b + X @ Wr^T, BN, ReLU
//                             using V_WMMA_F32_16X16X4_F32 (fp32 matrix path).
// ---------------------------------------------------------------------------

typedef __attribute__((ext_vector_type(2))) float v2f;
typedef __attribute__((ext_vector_type(8))) float v8f;

#define NNODES 100000
#define NEDGES 1600000
#define BN_EPS 1e-5f

// ---------------------------------------------------------------- degree ----
__global__ __launch_bounds__(256) void deg_kernel(const long long* __restrict__ ei,
                                                  float* __restrict__ deg) {
  int e = blockIdx.x * blockDim.x + threadIdx.x;
  if (e < NEDGES) {
    int dst = (int)ei[NEDGES + e];
    unsafeAtomicAdd(&deg[dst], 1.0f);
  }
}

// ---------------------------------------------------- scatter-add (E x D) ---
// One thread per (edge, 4-feature chunk): float4 gather + 4 FP32 atomics.
template <int D>
__global__ __launch_bounds__(256) void scatter_kernel(const float* __restrict__ x,
                                                      const long long* __restrict__ ei,
                                                      float* __restrict__ agg) {
  constexpr int C = D / 4;  // chunks per edge (power of two)
  long long tid = (long long)blockIdx.x * blockDim.x + threadIdx.x;
  if (tid >= (long long)NEDGES * C) return;
  int e = (int)(tid / C);
  int c = ((int)tid & (C - 1)) * 4;
  int src = (int)ei[e];
  int dst = (int)ei[NEDGES + e];
  const float4 v = *(const float4*)(x + (long long)src * D + c);
  float* p = agg + (long long)dst * D + c;
  unsafeAtomicAdd(p + 0, v.x);
  unsafeAtomicAdd(p + 1, v.y);
  unsafeAtomicAdd(p + 2, v.z);
  unsafeAtomicAdd(p + 3, v.w);
}

// ------------------------------------------------- fused WMMA SAGE layer ----
// Block = 128 threads (4 wave32), one block per 16-node row tile.
// Each wave produces 16x16 output tiles: C = A_agg*Bl + A_x*Br, striding the
// DOUT/16 column tiles across the 4 waves. fp32 WMMA, K=4 per instruction.
//
// Fragment layouts (CDNA5 ISA 7.12.2, 32-bit A / B / C, wave32):
//   A 16x4 : lane l -> row m = l&15, holds K = {kh, kh+1}, kh = (l>=16)?2:0
//   B 4x16 : lane l -> col n = l&15, holds K = {kh, kh+1}
//   C 16x16: lane l -> col n = l&15, VGPR v -> row m = v + ((l>=16)?8:0)
template <int DIN, int DOUT, bool BN_RELU>
__global__ __launch_bounds__(128) void sage_wmma_kernel(
    const float* __restrict__ agg, const float* __restrict__ xin,
    const float* __restrict__ deg,
    const float* __restrict__ wl, const float* __restrict__ bl,
    const float* __restrict__ wr,
    const float* __restrict__ bng, const float* __restrict__ bnb,
    const float* __restrict__ bnm, const float* __restrict__ bnv,
    float* __restrict__ out) {
  const int lane = threadIdx.x & 31;
  const int wave = threadIdx.x >> 5;
  const int node0 = blockIdx.x * 16;
  const int m = lane & 15;            // A row / B col / C col served by lane
  const int kh = (lane >> 4) << 1;    // K-half offset: 0 or 2
  const int hi8 = (lane >> 4) << 3;   // C row offset: 0 or 8

  // mean normalization folded into A-fragment loads
  const float invd = 1.0f / fmaxf(deg[node0 + m], 1.0f);

  const float* aggRow = agg + (long long)(node0 + m) * DIN + kh;
  const float* xRow   = xin + (long long)(node0 + m) * DIN + kh;

  for (int ct = wave; ct < DOUT / 16; ct += 4) {
    const int col0 = ct * 16;
    const float* wlRow = wl + (long long)(col0 + m) * DIN + kh;  // B = Wl^T
    const float* wrRow = wr + (long long)(col0 + m) * DIN + kh;  // B = Wr^T

    v8f c = {};
#pragma unroll 8
    for (int k = 0; k < DIN; k += 4) {
      v2f a = *(const v2f*)(aggRow + k);
      a *= invd;
      v2f b = *(const v2f*)(wlRow + k);
      c = __builtin_amdgcn_wmma_f32_16x16x4_f32(false, a, false, b,
                                                (short)0, c, false, false);
    }
#pragma unroll 8
    for (int k = 0; k < DIN; k += 4) {
      v2f a = *(const v2f*)(xRow + k);
      v2f b = *(const v2f*)(wrRow + k);
      c = __builtin_amdgcn_wmma_f32_16x16x4_f32(false, a, false, b,
                                                (short)0, c, false, false);
    }

    // Epilogue: bias (+ BN folded to scale/shift) + ReLU, per output column.
    const int ncol = col0 + m;
    float scale = 1.0f, shift = bl[ncol];
    if (BN_RELU) {
      const float is = bng[ncol] * rsqrtf(bnv[ncol] + BN_EPS);
      shift = (shift - bnm[ncol]) * is + bnb[ncol];
      scale = is;
    }
    float* obase = out + (long long)(node0 + hi8) * DOUT + ncol;
#pragma unroll
    for (int v = 0; v < 8; ++v) {
      float y = c[v] * scale + shift;
      if (BN_RELU) y = fmaxf(y, 0.0f);
      obase[(long long)v * DOUT] = y;
    }
  }
}

// ---------------------------------------------------------------------------
extern "C" void kernel_launch(void* const* d_in, const int* in_sizes, int n_in,
                              void* d_out, int out_size, void* d_ws, size_t ws_size,
                              hipStream_t stream) {
  const float*     x    = (const float*)d_in[0];
  const long long* ei   = (const long long*)d_in[1];
  const float* w_l1 = (const float*)d_in[2];
  const float* b_l1 = (const float*)d_in[3];
  const float* w_r1 = (const float*)d_in[4];
  const float* bn1g = (const float*)d_in[5];
  const float* bn1b = (const float*)d_in[6];
  const float* bn1m = (const float*)d_in[7];
  const float* bn1v = (const float*)d_in[8];
  const float* w_l2 = (const float*)d_in[9];
  const float* b_l2 = (const float*)d_in[10];
  const float* w_r2 = (const float*)d_in[11];
  const float* bn2g = (const float*)d_in[12];
  const float* bn2b = (const float*)d_in[13];
  const float* bn2m = (const float*)d_in[14];
  const float* bn2v = (const float*)d_in[15];
  const float* w_l3 = (const float*)d_in[16];
  const float* b_l3 = (const float*)d_in[17];
  const float* w_r3 = (const float*)d_in[18];
  float* outp = (float*)d_out;

  // Workspace layout: deg[N] | agg[N*128] | h1[N*128] | h2[N*128]  (~154 MB)
  float* deg = (float*)d_ws;
  float* agg = deg + NNODES;
  float* h1  = agg + (size_t)NNODES * 128;
  float* h2  = h1  + (size_t)NNODES * 128;

  const size_t aggBytes = (size_t)NNODES * 128 * sizeof(float);
  const int rowBlocks = NNODES / 16;                       // 6250, exact
  const int scatBlocks = (NEDGES * (128 / 4)) / 256;       // 200000, exact
  const int degBlocks = (NEDGES + 255) / 256;

  hipMemsetAsync(deg, 0, NNODES * sizeof(float), stream);
  deg_kernel<<<degBlocks, 256, 0, stream>>>(ei, deg);

  // ---- layer 1: x -> h1 -----------------------------------------------
  hipMemsetAsync(agg, 0, aggBytes, stream);
  scatter_kernel<128><<<scatBlocks, 256, 0, stream>>>(x, ei, agg);
  sage_wmma_kernel<128, 128, true><<<rowBlocks, 128, 0, stream>>>(
      agg, x, deg, w_l1, b_l1, w_r1, bn1g, bn1b, bn1m, bn1v, h1);

  // ---- layer 2: h1 -> h2 ----------------------------------------------
  hipMemsetAsync(agg, 0, aggBytes, stream);
  scatter_kernel<128><<<scatBlocks, 256, 0, stream>>>(h1, ei, agg);
  sage_wmma_kernel<128, 128, true><<<rowBlocks, 128, 0, stream>>>(
      agg, h1, deg, w_l2, b_l2, w_r2, bn2g, bn2b, bn2m, bn2v, h2);

  // ---- layer 3: h2 -> out (no BN / ReLU, DOUT=64) ---------------------
  hipMemsetAsync(agg, 0, aggBytes, stream);
  scatter_kernel<128><<<scatBlocks, 256, 0, stream>>>(h2, ei, agg);
  sage_wmma_kernel<128, 64, false><<<rowBlocks, 128, 0, stream>>>(
      agg, h2, deg, w_l3, b_l3, w_r3, nullptr, nullptr, nullptr, nullptr, outp);
}